// S4DKernel_19550691131556
// MI455X (gfx1250) — compile-verified
//
#include <hip/hip_runtime.h>
#include <hip/hip_bf16.h>
#include <math.h>

// S4D Vandermonde kernel for gfx1250 (MI455X).
// K[h, 64q+r] = 2*Re( sum_n (Ceff_n * z64_n^q) * (z_n^r) )
// => per-h real GEMM D(64x64) = [2*Ceff_re | -2*Ceff_im panel] (64x64) @ [B_re ; B_im] (64x64)
// executed with V_WMMA_F32_16X16X4_F32 (f32-exact matrix path).
// B is stored K-transposed in LDS so both A and B fragments are single
// 8-byte-aligned ds_load_b64 with immediate offsets (STRIDE=66 keeps all
// 64 banks conflict-free for the strided fragment reads).

typedef __attribute__((ext_vector_type(2))) float v2f;
typedef __attribute__((ext_vector_type(8))) float v8f;

#define HDIM   1024
#define NHALF  32
#define LLEN   4096
#define TSEG   64          // l = 64*q + r
#define STRIDE 66          // LDS row stride (dwords): even (8B pairs) + conflict-free

__global__ __launch_bounds__(128)
void s4d_vandermonde_wmma(const float* __restrict__ C,         // (H, 32, 2)
                          const float* __restrict__ log_dt,    // (H,)
                          const float* __restrict__ log_A_real,// (H, 32)
                          const float* __restrict__ A_imag,    // (H, 32)
                          float* __restrict__ out)             // (H, 4096)
{
    __shared__ __align__(16) float sA [TSEG * STRIDE];  // A panel: row q, cols [2*Re | -2*Im]
    __shared__ __align__(16) float sBt[TSEG * STRIDE];  // B^T panel: row r, cols [Re(z^r) | Im(z^r)]
    __shared__ float s_dre[NHALF], s_dim[NHALF];        // dtA
    __shared__ float s_d64re[NHALF], s_d64im[NHALF];    // 64*dtA
    __shared__ float s_cre[NHALF], s_cim[NHALF];        // 2*Ceff

    const int h   = blockIdx.x;
    const int tid = threadIdx.x;

    // ---------------- Phase 1: per-n scalars (32 lanes) ----------------
    if (tid < NHALF) {
        const int n = tid;
        const float dt  = __expf(log_dt[h]);
        const float Are = -__expf(log_A_real[h * NHALF + n]);
        const float Aim = A_imag[h * NHALF + n];
        const float dre = Are * dt;
        const float dim = Aim * dt;

        // z = exp(dtA)
        float sn, cs;
        __sincosf(dim, &sn, &cs);
        const float e  = __expf(dre);
        const float zr = e * cs;
        const float zi = e * sn;

        // Ceff = Cc * (z - 1) / A   (complex); fold in the final *2 here.
        const float numr = zr - 1.0f;
        const float numi = zi;
        const float inv_d = __frcp_rn(Are * Are + Aim * Aim);
        const float t1r = (numr * Are + numi * Aim) * inv_d;   // (z-1)*conj(A)/|A|^2
        const float t1i = (numi * Are - numr * Aim) * inv_d;
        const float ccr = C[(h * NHALF + n) * 2 + 0];
        const float cci = C[(h * NHALF + n) * 2 + 1];

        s_dre[n]   = dre;
        s_dim[n]   = dim;
        s_d64re[n] = 64.0f * dre;
        s_d64im[n] = 64.0f * dim;
        s_cre[n]   = 2.0f * (ccr * t1r - cci * t1i);
        s_cim[n]   = 2.0f * (ccr * t1i + cci * t1r);
    }
    __syncthreads();

    // ---------------- Phase 2: build operand panels in LDS ----------------
    // B^T: row r, col n      -> Re(z_n^r);  col 32+n -> Im(z_n^r)
    // A  : row q, col n      -> Re(2*Ceff_n * z64_n^q); col 32+n -> -Im(...)
    // Both fills have consecutive lanes writing consecutive LDS addresses.
    for (int idx = tid; idx < 2 * TSEG * NHALF; idx += 128) {
        if (idx < TSEG * NHALF) {            // B panel: n = idx&31, r = idx>>5
            const int n = idx & 31;
            const int r = idx >> 5;
            const float fr = (float)r;
            float sn, cs;
            __sincosf(fr * s_dim[n], &sn, &cs);
            const float er = __expf(fr * s_dre[n]);
            sBt[r * STRIDE + n]         = er * cs;   // Re(z^r)
            sBt[r * STRIDE + n + NHALF] = er * sn;   // Im(z^r)
        } else {                             // A panel: i = q*32 + n
            const int i = idx - TSEG * NHALF;
            const int n = i & 31;
            const int q = i >> 5;
            const float fq = (float)q;
            float sn, cs;
            __sincosf(fq * s_d64im[n], &sn, &cs);
            const float er = __expf(fq * s_d64re[n]);
            const float wr = er * cs;                 // Re(z64^q)
            const float wi = er * sn;                 // Im(z64^q)
            sA[q * STRIDE + n]         =  s_cre[n] * wr - s_cim[n] * wi;   //  2*Re
            sA[q * STRIDE + n + NHALF] = -(s_cre[n] * wi + s_cim[n] * wr); // -2*Im
        }
    }
    __syncthreads();

    // ---------------- Phase 3: 64x64x64 GEMM via v_wmma_f32_16x16x4_f32 ----
    const int wave = tid >> 5;          // wave32: 4 waves
    const int lane = tid & 31;
    const int half = lane >> 4;         // fragment half (lanes 0-15 vs 16-31)
    const int lid  = lane & 15;
    const int q0   = wave * 16;         // this wave's output row stripe

    // Per-lane fragment base pointers: all K-step loads are single aligned
    // b64 loads with small immediate offsets from these bases.
    const v2f* __restrict__ pA = (const v2f*)(sA + (q0 + lid) * STRIDE);  // + ka/2
    float* __restrict__ orow = out + (size_t)h * LLEN;

    #pragma unroll
    for (int c4 = 0; c4 < 4; ++c4) {
        const int r0 = c4 * 16;
        const v2f* __restrict__ pB = (const v2f*)(sBt + (r0 + lid) * STRIDE);
        v8f acc = {};
        #pragma unroll
        for (int k0 = 0; k0 < TSEG; k0 += 4) {
            const int ka = k0 + half * 2;   // lanes 0-15: K=k0,k0+1 ; lanes 16-31: K=k0+2,k0+3
            const v2f a = pA[ka >> 1];      // (A[q, ka], A[q, ka+1])
            const v2f b = pB[ka >> 1];      // (B[ka, r], B[ka+1, r]) from B^T row
            acc = __builtin_amdgcn_wmma_f32_16x16x4_f32(
                false, a, false, b, (short)0, acc, false, false);
        }
        // C/D layout: VGPR j -> (M = q0 + j + half*8, N = r0 + lid)
        #pragma unroll
        for (int j = 0; j < 8; ++j) {
            const int m = q0 + j + half * 8;
            orow[m * TSEG + r0 + lid] = acc[j];
        }
    }
}

extern "C" void kernel_launch(void* const* d_in, const int* in_sizes, int n_in,
                              void* d_out, int out_size, void* d_ws, size_t ws_size,
                              hipStream_t stream) {
    const float* C          = (const float*)d_in[0];
    const float* log_dt     = (const float*)d_in[1];
    const float* log_A_real = (const float*)d_in[2];
    const float* A_imag     = (const float*)d_in[3];
    // d_in[4] is L (== 4096), compile-time constant here.
    float* out = (float*)d_out;

    s4d_vandermonde_wmma<<<HDIM, 128, 0, stream>>>(C, log_dt, log_A_real, A_imag, out);
}